// Lap_GCN_18107582120780
// MI455X (gfx1250) — compile-verified
//
#include <hip/hip_runtime.h>
#include <hip/hip_bf16.h>

// ---------------------------------------------------------------------------
// Lap-GCN forward for MI455X (gfx1250, wave32).
//   - Dense GEMMs via V_WMMA_F32_16X16X4_F32 (f32 in, f32 accum == reference).
//   - A/B tile staging via GLOBAL_LOAD_ASYNC_TO_LDS_B128 (ASYNCcnt-tracked,
//     no VGPR round-trip), waited with s_wait_asynccnt before the barrier.
//   - Edge aggregation: h (12.8MB) is L2-resident (192MB L2); scatter uses
//     native global_atomic_add_f32 via unsafeAtomicAdd.
// ---------------------------------------------------------------------------

#define NNODES 50000
#define NPAD   50016          // round up to multiple of 32 (GEMM row tiles)
#define NEDGE  1250000
#define NFEAT  128
#define NHID   64
#define NCLS   47
#define NLAY   4

typedef __attribute__((ext_vector_type(2))) float v2f;
typedef __attribute__((ext_vector_type(8))) float v8f;

// Async 16-byte copy global -> LDS (GVS mode: saddr base + 32-bit voffset).
// LDS dest address = low 32 bits of the flat shared pointer (aperture keeps
// the LDS offset in addr[31:0] per CDNA5 ISA 10.2).
__device__ __forceinline__ void async_copy16(float* lds_dst, const float* base,
                                             int byte_off) {
  uint32_t lds_off = (uint32_t)(uintptr_t)lds_dst;
  asm volatile("global_load_async_to_lds_b128 %0, %1, %2"
               :: "v"(lds_off), "v"(byte_off), "s"(base)
               : "memory");
}

__device__ __forceinline__ void async_wait_all() {
  asm volatile("s_wait_asynccnt 0" ::: "memory");
}

// One 16x16 output tile per wave, K-loop of v_wmma_f32_16x16x4_f32.
// A fragment layout (16x4 f32): lanes 0-15 hold row M=lane, K = {0,1};
// lanes 16-31 hold row M=lane-16, K = {2,3}. B mirrors (rows K, cols N=lane).
__device__ __forceinline__ v8f wmma_tile_f32(const float* ldsA, const float* ldsB,
                                             int mloc, int n0, int K,
                                             int ASTR, int BSTR, int lane) {
  const int half = lane >> 4;      // which K-pair this lane holds
  const int lid  = lane & 15;
  const float* aRow = ldsA + (mloc + lid) * ASTR;
  v8f c = {0.f, 0.f, 0.f, 0.f, 0.f, 0.f, 0.f, 0.f};
  for (int k = 0; k < K; k += 4) {
    const int ka = k + 2 * half;
    v2f a, b;
    a.x = aRow[ka];
    a.y = aRow[ka + 1];
    b.x = ldsB[ka * BSTR + n0 + lid];
    b.y = ldsB[(ka + 1) * BSTR + n0 + lid];
    c = __builtin_amdgcn_wmma_f32_16x16x4_f32(false, a, false, b,
                                              (short)0, c, false, false);
  }
  return c;
}

// ------------------------------ degree / dinv ------------------------------

__global__ void k_deg_init(float* deg) {
  int i = blockIdx.x * blockDim.x + threadIdx.x;
  if (i < NPAD) deg[i] = (i < NNODES) ? 1.0f : 0.0f;  // self-loop counts 1
}

__global__ void k_deg_edges(const int* __restrict__ col, float* deg) {
  int e = blockIdx.x * blockDim.x + threadIdx.x;
  if (e < NEDGE) unsafeAtomicAdd(&deg[col[e]], 1.0f);
}

__global__ void k_dinv(float* deg) {
  int i = blockIdx.x * blockDim.x + threadIdx.x;
  if (i < NPAD) {
    float d = deg[i];
    deg[i] = (d > 0.0f) ? rsqrtf(d) : 0.0f;
  }
}

// ------------------------------ encoder GEMM -------------------------------
// X = relu(x @ enc_w + enc_b), M=NPAD, K=128, N=64.  Block = 8 waves = 32x64.

__global__ void k_encoder(const float* __restrict__ x,
                          const float* __restrict__ enc_w,
                          const float* __restrict__ enc_b,
                          float* __restrict__ X) {
  const int ASTR = NFEAT + 4;   // 132: row bank-advance of 4 -> conflict-free
  const int BSTR = NHID + 16;   // 80: 2-row offset = 32 banks -> halves disjoint
  __shared__ float ldsA[32 * (NFEAT + 4)];
  __shared__ float ldsB[NFEAT * (NHID + 16)];
  const int tid  = threadIdx.x;
  const int row0 = blockIdx.x * 32;

  // A: 32 rows x 32 16B-chunks, rows clamped for the padded tail
  for (int idx = tid; idx < 32 * 32; idx += 256) {
    int r = idx >> 5, c4 = (idx & 31) << 2;
    int gr = row0 + r; if (gr >= NNODES) gr = NNODES - 1;
    async_copy16(&ldsA[r * ASTR + c4], x, (gr * NFEAT + c4) * 4);
  }
  // B: 128 rows x 16 chunks
  for (int idx = tid; idx < NFEAT * 16; idx += 256) {
    int r = idx >> 4, c4 = (idx & 15) << 2;
    async_copy16(&ldsB[r * BSTR + c4], enc_w, (r * NHID + c4) * 4);
  }
  async_wait_all();
  __syncthreads();

  const int wave = tid >> 5, lane = tid & 31;
  const int mloc = (wave & 1) * 16;
  const int n0   = (wave >> 1) * 16;
  v8f acc = wmma_tile_f32(ldsA, ldsB, mloc, n0, NFEAT, ASTR, BSTR, lane);

  const int half = lane >> 4, lid = lane & 15;
  const int gn    = n0 + lid;
  const float b   = enc_b[gn];
  const int rbase = row0 + mloc + half * 8;
#pragma unroll
  for (int r = 0; r < 8; ++r) {
    float v = acc[r] + b;
    X[(size_t)(rbase + r) * NHID + gn] = v > 0.f ? v : 0.f;
  }
}

// --------------------------- h = X @ conv_w --------------------------------

__global__ void k_gemm_h(const float* __restrict__ X,
                         const float* __restrict__ conv_w,
                         float* __restrict__ h) {
  const int ASTR = NHID + 4;    // 68 (row = 272B, 16B aligned)
  const int BSTR = NHID + 16;   // 80 (row = 320B, 16B aligned)
  __shared__ float ldsA[32 * (NHID + 4)];
  __shared__ float ldsB[NHID * (NHID + 16)];
  const int tid  = threadIdx.x;
  const int row0 = blockIdx.x * 32;

  for (int idx = tid; idx < 32 * 16; idx += 256) {       // A: 32 x 16 chunks
    int r = idx >> 4, c4 = (idx & 15) << 2;
    async_copy16(&ldsA[r * ASTR + c4], X, ((row0 + r) * NHID + c4) * 4);
  }
  for (int idx = tid; idx < NHID * 16; idx += 256) {     // B: 64 x 16 chunks
    int r = idx >> 4, c4 = (idx & 15) << 2;
    async_copy16(&ldsB[r * BSTR + c4], conv_w, (r * NHID + c4) * 4);
  }
  async_wait_all();
  __syncthreads();

  const int wave = tid >> 5, lane = tid & 31;
  const int mloc = (wave & 1) * 16;
  const int n0   = (wave >> 1) * 16;
  v8f acc = wmma_tile_f32(ldsA, ldsB, mloc, n0, NHID, ASTR, BSTR, lane);

  const int half = lane >> 4, lid = lane & 15;
  const int gn    = n0 + lid;
  const int rbase = row0 + mloc + half * 8;
#pragma unroll
  for (int r = 0; r < 8; ++r)
    h[(size_t)(rbase + r) * NHID + gn] = acc[r];
}

// ---------------- aggregation: agg = conv_b + D^-1/2 (A+I) D^-1/2 h --------

__global__ void k_agg_init(const float* __restrict__ h,
                           const float* __restrict__ conv_b,
                           const float* __restrict__ dinv,
                           float* __restrict__ agg) {
  int idx = blockIdx.x * blockDim.x + threadIdx.x;
  if (idx < NPAD * NHID) {
    int v = idx >> 6, f = idx & (NHID - 1);
    float di = dinv[v];
    agg[idx] = conv_b[f] + di * di * h[idx];     // self-loop term
  }
}

__global__ void k_agg_edges(const int* __restrict__ row,
                            const int* __restrict__ col,
                            const float* __restrict__ dinv,
                            const float* __restrict__ h,
                            float* __restrict__ agg) {
  long tid = (long)blockIdx.x * blockDim.x + threadIdx.x;
  if (tid >= (long)NEDGE * 16) return;
  int e = (int)(tid >> 4);
  int q = ((int)tid & 15) << 2;                  // feature quad
  int r = row[e], c = col[e];
  float nrm = dinv[r] * dinv[c];
  const float4 hv = *(const float4*)(h + (size_t)r * NHID + q);
  float* dst = agg + (size_t)c * NHID + q;
  unsafeAtomicAdd(dst + 0, nrm * hv.x);
  unsafeAtomicAdd(dst + 1, nrm * hv.y);
  unsafeAtomicAdd(dst + 2, nrm * hv.z);
  unsafeAtomicAdd(dst + 3, nrm * hv.w);
}

// ---- fused: res GEMM + gate:  X = tanh(eps)*X + relu(agg - h@res_w - res_b)

__global__ void k_res_update(const float* __restrict__ h,
                             const float* __restrict__ res_w,
                             const float* __restrict__ res_b,
                             const float* __restrict__ agg,
                             const float* __restrict__ eps,   // this layer's row
                             float* __restrict__ X) {
  const int ASTR = NHID + 4;
  const int BSTR = NHID + 16;
  __shared__ float ldsA[32 * (NHID + 4)];
  __shared__ float ldsB[NHID * (NHID + 16)];
  const int tid  = threadIdx.x;
  const int row0 = blockIdx.x * 32;

  for (int idx = tid; idx < 32 * 16; idx += 256) {
    int r = idx >> 4, c4 = (idx & 15) << 2;
    async_copy16(&ldsA[r * ASTR + c4], h, ((row0 + r) * NHID + c4) * 4);
  }
  for (int idx = tid; idx < NHID * 16; idx += 256) {
    int r = idx >> 4, c4 = (idx & 15) << 2;
    async_copy16(&ldsB[r * BSTR + c4], res_w, (r * NHID + c4) * 4);
  }
  async_wait_all();
  __syncthreads();

  const int wave = tid >> 5, lane = tid & 31;
  const int mloc = (wave & 1) * 16;
  const int n0   = (wave >> 1) * 16;
  v8f acc = wmma_tile_f32(ldsA, ldsB, mloc, n0, NHID, ASTR, BSTR, lane);

  const int half = lane >> 4, lid = lane & 15;
  const int gn    = n0 + lid;
  const float rb  = res_b[gn];
  const float te  = tanhf(eps[gn]);              // coeff-1 (X0 == X always)
  const int rbase = row0 + mloc + half * 8;
#pragma unroll
  for (int r = 0; r < 8; ++r) {
    size_t o = (size_t)(rbase + r) * NHID + gn;
    float v = agg[o] - acc[r] - rb;              // conv_out + res
    v = v > 0.f ? v : 0.f;
    X[o] = te * X[o] + v;                        // DT=1, ALPHA=1
  }
}

// ------------------------------ decoder GEMM -------------------------------
// out = X @ dec_w + dec_b, N columns = 47 (zero-padded to 48 in LDS, so this
// kernel keeps plain VGPR staging -- async copies cannot pad).
// Block = 96 threads = 3 waves: one 16-row slab x three 16-col tiles.

__global__ void k_decoder(const float* __restrict__ X,
                          const float* __restrict__ dec_w,
                          const float* __restrict__ dec_b,
                          float* __restrict__ out) {
  const int ASTR = NHID + 4;    // 68
  const int BSTR = 48 + 16;     // 64 cols padded to 80-byte-safe stride
  __shared__ float ldsA[16 * (NHID + 4)];
  __shared__ float ldsB[NHID * (48 + 16)];
  const int tid  = threadIdx.x;
  const int row0 = blockIdx.x * 16;

  for (int idx = tid; idx < 16 * NHID; idx += 96) {
    int r = idx >> 6, c = idx & (NHID - 1);
    ldsA[r * ASTR + c] = X[(size_t)(row0 + r) * NHID + c];
  }
  for (int idx = tid; idx < NHID * 48; idx += 96) {
    int r = idx / 48, c = idx % 48;
    ldsB[r * BSTR + c] = (c < NCLS) ? dec_w[r * NCLS + c] : 0.0f;
  }
  __syncthreads();

  const int wave = tid >> 5, lane = tid & 31;
  const int n0 = wave * 16;
  v8f acc = wmma_tile_f32(ldsA, ldsB, 0, n0, NHID, ASTR, BSTR, lane);

  const int half = lane >> 4, lid = lane & 15;
  const int gn = n0 + lid;
  const float b = (gn < NCLS) ? dec_b[gn] : 0.0f;
  const int rbase = row0 + half * 8;
#pragma unroll
  for (int r = 0; r < 8; ++r) {
    int grow = rbase + r;
    if (grow < NNODES && gn < NCLS)
      out[(size_t)grow * NCLS + gn] = acc[r] + b;
  }
}

// ------------------------------- launcher ----------------------------------

extern "C" void kernel_launch(void* const* d_in, const int* in_sizes, int n_in,
                              void* d_out, int out_size, void* d_ws, size_t ws_size,
                              hipStream_t stream) {
  const float* x      = (const float*)d_in[0];
  const int*   edges  = (const int*)  d_in[1];   // (2, E): row then col
  const float* enc_w  = (const float*)d_in[2];
  const float* enc_b  = (const float*)d_in[3];
  const float* conv_w = (const float*)d_in[4];
  const float* conv_b = (const float*)d_in[5];
  const float* res_w  = (const float*)d_in[6];
  const float* res_b  = (const float*)d_in[7];
  const float* dec_w  = (const float*)d_in[8];
  const float* dec_b  = (const float*)d_in[9];
  const float* eps    = (const float*)d_in[10];  // (NLAY, NHID)
  float* out = (float*)d_out;

  const int* erow = edges;
  const int* ecol = edges + NEDGE;

  float* ws   = (float*)d_ws;
  float* dinv = ws;                                  // NPAD (deg, then rsqrt)
  float* X    = dinv + NPAD;                         // NPAD x 64
  float* h    = X + (size_t)NPAD * NHID;             // NPAD x 64
  float* agg  = h + (size_t)NPAD * NHID;             // NPAD x 64

  // normalization  D^{-1/2}
  k_deg_init <<<(NPAD + 255) / 256, 256, 0, stream>>>(dinv);
  k_deg_edges<<<(NEDGE + 255) / 256, 256, 0, stream>>>(ecol, dinv);
  k_dinv     <<<(NPAD + 255) / 256, 256, 0, stream>>>(dinv);

  // encoder
  k_encoder<<<NPAD / 32, 256, 0, stream>>>(x, enc_w, enc_b, X);

  // 4 GCN layers
  for (int i = 0; i < NLAY; ++i) {
    k_gemm_h   <<<NPAD / 32, 256, 0, stream>>>(X, conv_w, h);
    k_agg_init <<<(NPAD * NHID + 255) / 256, 256, 0, stream>>>(h, conv_b, dinv, agg);
    k_agg_edges<<<(int)(((long)NEDGE * 16 + 255) / 256), 256, 0, stream>>>(
        erow, ecol, dinv, h, agg);
    k_res_update<<<NPAD / 32, 256, 0, stream>>>(h, res_w, res_b, agg,
                                                eps + i * NHID, X);
  }

  // decoder
  k_decoder<<<NPAD / 16, 96, 0, stream>>>(X, dec_w, dec_b, out);
}